// PointsNonMaxSuppression_90855738179815
// MI455X (gfx1250) — compile-verified
//
#include <hip/hip_runtime.h>
#include <stdint.h>

// Problem shape (fixed by reference): [B=16, C=80, H=128, W=128] f32
#define NMS_B 16
#define NMS_C 80
#define NMS_H 128
#define NMS_W 128
#define PLANE (NMS_H * NMS_W)   // 16384 floats = 64 KB

// One workgroup per (b,c) plane. 256 threads = 8 wave32.
// Stage 1: async-copy the whole 64KB plane global -> LDS
//          (global_load_async_to_lds_b128, ASYNCcnt, s_wait_asynccnt).
// Stage 2: thread t owns column x = t&127, rows [ (t>>7)*64, +64 ).
//          Rolling 3x3 max out of LDS, coalesced b32 stores.
__global__ __launch_bounds__(256) void
PointsNMS_gfx1250_kernel(const float* __restrict__ in, float* __restrict__ out)
{
    __shared__ float tile[PLANE];   // 65536 bytes

    const int plane = blockIdx.x;                       // 0..1279
    const float* __restrict__ src = in  + (size_t)plane * PLANE;
    float*       __restrict__ dst = out + (size_t)plane * PLANE;

    const int t = threadIdx.x;                          // 0..255

    // ---------------- Stage 1: async bulk copy to LDS ----------------
    {
        // LDS byte offset of tile[] : generic (flat) address of a __shared__
        // object carries the LDS byte offset in its low 32 bits.
        const uint32_t lds_base = (uint32_t)(uintptr_t)(&tile[0]);
        #pragma unroll
        for (int i = 0; i < 16; ++i) {
            const int elem = (t + i * 256) * 4;              // float index, lane-consecutive 16B
            const uint64_t gaddr = (uint64_t)(uintptr_t)(src + elem);
            const uint32_t laddr = lds_base + (uint32_t)elem * 4u;
            // VDST = LDS byte address, VADDR = 64-bit global address (GV mode)
            asm volatile("global_load_async_to_lds_b128 %0, %1, off"
                         :: "v"(laddr), "v"(gaddr)
                         : "memory");
        }
        // Wait for this wave's async transfers, then sync the workgroup so
        // every wave sees the full plane in LDS.
        asm volatile("s_wait_asynccnt 0" ::: "memory");
    }
    __syncthreads();

    // ---------------- Stage 2: rolling 3x3 max ----------------
    const int  x   = t & (NMS_W - 1);      // column, lane-consecutive
    const int  y0  = (t >> 7) * 64;        // row strip start (0 or 64), wave-uniform
    const float NINF = -__builtin_inff();
    const bool has_l = (x > 0);
    const bool has_r = (x < NMS_W - 1);

    // h(y) = max over columns x-1..x+1 of row y; also returns center value.
    auto hrow = [&](int y, float& center) -> float {
        const float* row = &tile[y * NMS_W + x];
        const float c = row[0];
        const float l = has_l ? row[-1] : NINF;
        const float r = has_r ? row[ 1] : NINF;
        center = c;
        return fmaxf(fmaxf(l, c), r);
    };

    float ccur, cnext, cdead;
    float hprev = (y0 > 0) ? hrow(y0 - 1, cdead) : NINF;   // wave-uniform branch
    float hcur  = hrow(y0, ccur);

    #pragma unroll 4
    for (int yy = 0; yy < 64; ++yy) {
        const int y = y0 + yy;
        float hnext;
        if (y + 1 < NMS_H) {            // wave-uniform (only last row of strip 1)
            hnext = hrow(y + 1, cnext);
        } else {
            hnext = NINF;
            cnext = NINF;
        }
        const float m = fmaxf(fmaxf(hprev, hcur), hnext);
        const float p = ccur;
        dst[y * NMS_W + x] = (p == m) ? p : 0.0f;          // coalesced b32 store
        hprev = hcur;
        hcur  = hnext;
        ccur  = cnext;
    }
}

extern "C" void kernel_launch(void* const* d_in, const int* in_sizes, int n_in,
                              void* d_out, int out_size, void* d_ws, size_t ws_size,
                              hipStream_t stream)
{
    (void)in_sizes; (void)n_in; (void)out_size; (void)d_ws; (void)ws_size;
    const float* in  = (const float*)d_in[0];
    float*       out = (float*)d_out;

    const dim3 grid(NMS_B * NMS_C);   // 1280 planes
    const dim3 block(256);
    PointsNMS_gfx1250_kernel<<<grid, block, 0, stream>>>(in, out);
}